// Self_Att_Bundle_Freq_weight_87488483819795
// MI455X (gfx1250) — compile-verified
//
#include <hip/hip_runtime.h>
#include <hip/hip_bf16.h>
#include <math.h>

// ---------------------------------------------------------------------------
// Problem constants (from reference)
// ---------------------------------------------------------------------------
#define BATCH 32
#define TLEN  1600
#define FDIM  200
#define NPOOL 20          // pooled T dim
#define BT    80          // TLEN / NPOOL
#define PF    100         // pooled F dim
#define DMODEL 256
#define HID   200         // TLEN/8

typedef __attribute__((ext_vector_type(2))) float v2f;
typedef __attribute__((ext_vector_type(8))) float v8f;

// ---------------------------------------------------------------------------
// K1: block pooling.  One block per (b, i) pooled row.
//   bsum[b,i,f] = sum_{r<80} x[b, i*80+r, f]          (B,20,200)
//   xp[b,i,c]   = (bsum[2c]+bsum[2c+1]) / 160         (B,20,100)
// ---------------------------------------------------------------------------
__global__ __launch_bounds__(256) void pool_kernel(const float* __restrict__ x,
                                                   float* __restrict__ xp,
                                                   float* __restrict__ bsum) {
    __shared__ float s_bs[FDIM];
    const int blk = blockIdx.x;                 // b*20 + i
    const int b = blk / NPOOL, i = blk % NPOOL;
    const int tid = threadIdx.x;
    const float* base = x + ((size_t)b * TLEN + (size_t)i * BT) * FDIM;
    if (tid < FDIM) {
        float acc = 0.f;
        #pragma unroll 4
        for (int r = 0; r < BT; ++r) acc += base[(size_t)r * FDIM + tid];
        s_bs[tid] = acc;
        bsum[(size_t)blk * FDIM + tid] = acc;
    }
    __syncthreads();
    if (tid < PF) {
        xp[(size_t)blk * PF + tid] = (s_bs[2 * tid] + s_bs[2 * tid + 1]) * (1.0f / 160.0f);
    }
}

// ---------------------------------------------------------------------------
// K2: per-batch attention.  One block per batch element.
//   k = xp@Wk+bk, q = xp@Wq+bq  (20x256 each)
//   att = softmax(k q^T / 16) over j
//   y[i,f] = sum_j att[i,j] * bsum[j,f];  emit yavg[i]=mean_f, ymax[i]=max_f
// ---------------------------------------------------------------------------
__global__ __launch_bounds__(256) void att_kernel(const float* __restrict__ xp,
                                                  const float* __restrict__ bsum,
                                                  const float* __restrict__ Wk,
                                                  const float* __restrict__ bk,
                                                  const float* __restrict__ Wq,
                                                  const float* __restrict__ bq,
                                                  float* __restrict__ yavg,
                                                  float* __restrict__ ymax) {
    __shared__ float s_xp[NPOOL * PF];        //  8 KB
    __shared__ float s_k[NPOOL * DMODEL];     // 20 KB
    __shared__ float s_q[NPOOL * DMODEL];     // 20 KB
    __shared__ float s_bsum[NPOOL * FDIM];    // 16 KB
    __shared__ float s_sc[NPOOL * NPOOL];
    __shared__ float s_att[NPOOL * NPOOL];

    const int b = blockIdx.x;
    const int tid = threadIdx.x;

    for (int o = tid; o < NPOOL * PF; o += 256) s_xp[o] = xp[(size_t)b * NPOOL * PF + o];
    for (int o = tid; o < NPOOL * FDIM; o += 256) s_bsum[o] = bsum[(size_t)b * NPOOL * FDIM + o];
    __syncthreads();

    // k and q projections
    for (int o = tid; o < NPOOL * DMODEL; o += 256) {
        const int i = o >> 8, d = o & 255;
        float ak = bk[d], aq = bq[d];
        for (int c = 0; c < PF; ++c) {
            const float xv = s_xp[i * PF + c];
            ak = fmaf(xv, Wk[c * DMODEL + d], ak);
            aq = fmaf(xv, Wq[c * DMODEL + d], aq);
        }
        s_k[o] = ak; s_q[o] = aq;
    }
    __syncthreads();

    // scores
    for (int o = tid; o < NPOOL * NPOOL; o += 256) {
        const int i = o / NPOOL, j = o % NPOOL;
        float s = 0.f;
        for (int d = 0; d < DMODEL; ++d) s = fmaf(s_k[i * DMODEL + d], s_q[j * DMODEL + d], s);
        s_sc[o] = s * (1.0f / 16.0f);
    }
    __syncthreads();

    // softmax over j (one thread per row)
    if (tid < NPOOL) {
        float mx = -1e30f;
        for (int j = 0; j < NPOOL; ++j) mx = fmaxf(mx, s_sc[tid * NPOOL + j]);
        float sum = 0.f;
        for (int j = 0; j < NPOOL; ++j) {
            const float e = __expf(s_sc[tid * NPOOL + j] - mx);
            s_att[tid * NPOOL + j] = e;
            sum += e;
        }
        const float inv = 1.0f / sum;
        for (int j = 0; j < NPOOL; ++j) s_att[tid * NPOOL + j] *= inv;
    }
    __syncthreads();

    // y row statistics (one thread per pooled row)
    if (tid < NPOOL) {
        float su = 0.f, mx = -1e30f;
        for (int f = 0; f < FDIM; ++f) {
            float v = 0.f;
            for (int j = 0; j < NPOOL; ++j) v = fmaf(s_att[tid * NPOOL + j], s_bsum[j * FDIM + f], v);
            su += v;
            mx = fmaxf(mx, v);
        }
        yavg[b * NPOOL + tid] = su * (1.0f / (float)FDIM);
        ymax[b * NPOOL + tid] = mx;
    }
}

// ---------------------------------------------------------------------------
// K3a: W1 block-row-sum:  W1bs[j,c] = sum_{r<80} W1[j*80+r, c]   (20,200)
// ---------------------------------------------------------------------------
__global__ __launch_bounds__(256) void w1bs_kernel(const float* __restrict__ W1,
                                                   float* __restrict__ W1bs) {
    const int o = blockIdx.x * 256 + threadIdx.x;
    if (o >= NPOOL * HID) return;
    const int j = o / HID, c = o % HID;
    float acc = 0.f;
    const float* base = W1 + ((size_t)j * BT) * HID + c;
    #pragma unroll 4
    for (int r = 0; r < BT; ++r) acc += base[(size_t)r * HID];
    W1bs[o] = acc;
}

// ---------------------------------------------------------------------------
// K3b: gate vector  g[b,c] = relu(yavg[b]@W1bs + b1)[c] + relu(ymax[b]@W1bs + b1)[c]
// ---------------------------------------------------------------------------
__global__ __launch_bounds__(256) void gate_kernel(const float* __restrict__ yavg,
                                                   const float* __restrict__ ymax,
                                                   const float* __restrict__ W1bs,
                                                   const float* __restrict__ b1,
                                                   float* __restrict__ g) {
    const int o = blockIdx.x * 256 + threadIdx.x;
    if (o >= BATCH * HID) return;
    const int b = o / HID, c = o % HID;
    float ha = b1[c], hm = b1[c];
    for (int j = 0; j < NPOOL; ++j) {
        const float w = W1bs[j * HID + c];
        ha = fmaf(yavg[b * NPOOL + j], w, ha);
        hm = fmaf(ymax[b * NPOOL + j], w, hm);
    }
    g[o] = fmaxf(ha, 0.f) + fmaxf(hm, 0.f);
}

// ---------------------------------------------------------------------------
// K4: WMMA GEMM + sigmoid epilogue.
//   scale(32,1600) = sigmoid( g(32,200) @ W2(200,1600) + 2*b2 )
// One wave (32 threads) per 16x16 output tile, v_wmma_f32_16x16x4_f32 over K.
//   A 16x4 frag: lanes 0-15 row M=lane {K=k,k+1}; lanes 16-31 row M=lane-16 {K=k+2,k+3}
//   B 4x16 frag: lanes 0-15 col N=lane rows {K=k,k+1}; lanes 16-31 rows {K=k+2,k+3}
//   C/D: VGPR i -> M = i + 8*(lane>=16), N = lane&15 (within tile)
// ---------------------------------------------------------------------------
__global__ __launch_bounds__(32) void gemm_sigmoid_kernel(const float* __restrict__ g,
                                                          const float* __restrict__ W2,
                                                          const float* __restrict__ b2,
                                                          float* __restrict__ scale) {
    const int lane = threadIdx.x;
    const int nt = blockIdx.x;          // 0..99  (N tiles of 16 over 1600)
    const int mt = blockIdx.y;          // 0..1   (M tiles of 16 over 32)
    const int half = lane >> 4;         // 0 | 1
    const int l15 = lane & 15;
    const int n0 = nt * 16;
    const int mrow = mt * 16 + l15;

    v8f acc = {};
    #pragma unroll 2
    for (int k = 0; k < HID; k += 4) {
        const int ka = k + half * 2;
        v2f a, bfr;
        a.x   = g[mrow * HID + ka];
        a.y   = g[mrow * HID + ka + 1];
        bfr.x = W2[(size_t)ka * TLEN + n0 + l15];
        bfr.y = W2[(size_t)(ka + 1) * TLEN + n0 + l15];
        acc = __builtin_amdgcn_wmma_f32_16x16x4_f32(
            /*neg_a=*/false, a, /*neg_b=*/false, bfr,
            /*c_mod=*/(short)0, acc, /*reuse_a=*/false, /*reuse_b=*/false);
    }

    const int n = n0 + l15;
    const float bias2 = 2.0f * b2[n];
    #pragma unroll
    for (int i = 0; i < 8; ++i) {
        const int m = mt * 16 + i + half * 8;
        const float v = acc[i] + bias2;
        scale[(size_t)m * TLEN + n] = 1.0f / (1.0f + __expf(-v));
    }
}

// ---------------------------------------------------------------------------
// K5: streaming apply  out[b,t,f] = x[b,t,f] * scale[b,t]   (float4 / b128)
// ---------------------------------------------------------------------------
__global__ __launch_bounds__(256) void apply_kernel(const float* __restrict__ x,
                                                    const float* __restrict__ scale,
                                                    float* __restrict__ out) {
    const size_t i4 = (size_t)blockIdx.x * 256 + threadIdx.x;   // float4 index
    const size_t total4 = (size_t)BATCH * TLEN * (FDIM / 4);
    if (i4 >= total4) return;
    const size_t row = i4 / (FDIM / 4);        // = b*TLEN + t  (FDIM%4==0, no row straddle)
    const float s = scale[row];
    float4 v = ((const float4*)x)[i4];
    v.x *= s; v.y *= s; v.z *= s; v.w *= s;
    ((float4*)out)[i4] = v;
}

// ---------------------------------------------------------------------------
// Launcher.  Inputs (setup_inputs order):
//   0:input(B,T,F) 1:Wk(100,256) 2:bk(256) 3:Wq(100,256) 4:bq(256)
//   5:W1(1600,200) 6:b1(200)     7:W2(200,1600) 8:b2(1600)
// ---------------------------------------------------------------------------
extern "C" void kernel_launch(void* const* d_in, const int* in_sizes, int n_in,
                              void* d_out, int out_size, void* d_ws, size_t ws_size,
                              hipStream_t stream) {
    const float* x  = (const float*)d_in[0];
    const float* Wk = (const float*)d_in[1];
    const float* bk = (const float*)d_in[2];
    const float* Wq = (const float*)d_in[3];
    const float* bq = (const float*)d_in[4];
    const float* W1 = (const float*)d_in[5];
    const float* b1 = (const float*)d_in[6];
    const float* W2 = (const float*)d_in[7];
    const float* b2 = (const float*)d_in[8];
    float* out = (float*)d_out;

    // workspace layout (floats)
    float* ws      = (float*)d_ws;
    float* xp      = ws;                                    // 32*20*100 = 64000
    float* bsum    = xp + BATCH * NPOOL * PF;               // 32*20*200 = 128000
    float* yavg    = bsum + BATCH * NPOOL * FDIM;           // 640
    float* ymax    = yavg + BATCH * NPOOL;                  // 640
    float* W1bs    = ymax + BATCH * NPOOL;                  // 4000
    float* gvec    = W1bs + NPOOL * HID;                    // 6400
    float* scale   = gvec + BATCH * HID;                    // 51200

    // K1: pooling (one block per pooled row)
    pool_kernel<<<BATCH * NPOOL, 256, 0, stream>>>(x, xp, bsum);

    // K2: attention + row stats (one block per batch)
    att_kernel<<<BATCH, 256, 0, stream>>>(xp, bsum, Wk, bk, Wq, bq, yavg, ymax);

    // K3a: W1 block sums
    w1bs_kernel<<<(NPOOL * HID + 255) / 256, 256, 0, stream>>>(W1, W1bs);

    // K3b: gate vector
    gate_kernel<<<(BATCH * HID + 255) / 256, 256, 0, stream>>>(yavg, ymax, W1bs, b1, gvec);

    // K4: WMMA GEMM + sigmoid -> scale(32,1600)
    {
        dim3 grid(TLEN / 16, BATCH / 16);   // (100, 2)
        gemm_sigmoid_kernel<<<grid, 32, 0, stream>>>(gvec, W2, b2, scale);
    }

    // K5: streaming out = x * scale
    {
        const size_t total4 = (size_t)BATCH * TLEN * (FDIM / 4);
        apply_kernel<<<(unsigned)((total4 + 255) / 256), 256, 0, stream>>>(x, scale, out);
    }
}